// RNN_27453430956109
// MI455X (gfx1250) — compile-verified
//
#include <hip/hip_runtime.h>

// out[bt,n] = alpha*relu(dot(u[bt,0:6], w_in[n,0:6]) + noise[bt,n]); rows with t==0 -> 0.
// Memory-bound: ~270MB traffic -> ~11.6us @ 23.3TB/s. The K=6 GEMM runs on
// V_WMMA_F32_16X16X4_F32 (two K=4 steps, K=6..7 zero-padded), keeping full fp32 precision
// and leaving VALU free for the fused relu/scale epilogue.

typedef float v2f __attribute__((ext_vector_type(2)));
typedef float v8f __attribute__((ext_vector_type(8)));

#define ALPHA_F 0.2f
#define T_DIM   1024
#define N_DIM   512
#define D_IN    6

// Pin a value in a VGPR so the compiler cannot sink its producing load under an
// exec-mask branch (keeps the row loads unconditional -> no divergent clauses).
__device__ __forceinline__ void pin_vgpr(float& a) { asm("" : "+v"(a)); }

__global__ __launch_bounds__(256) void rnn_wmma_kernel(
    const float* __restrict__ u,      // [B*T, 6]
    const float* __restrict__ w_in,   // [512, 6]
    const float* __restrict__ noise,  // [B*T, 512]
    float* __restrict__ out)          // [B*T, 512]
{
    const int lane = threadIdx.x & 31;
    const int wave = threadIdx.x >> 5;
    const int m    = lane & 15;           // M (for A) or N (for B/C) within tile
    const bool hi  = lane >= 16;

    // Block <-> one bt-tile (16 rows); wave <-> 4 of the 32 n-tiles.
    const int bt0 = blockIdx.x * 16;

    // ---- A operand: u rows bt0..bt0+15, K padded 6 -> 8 ----
    // Rows are 24B and 8B-aligned: three unconditional b64 loads, then cndmask selects.
    const float* urow = u + (long)(bt0 + m) * D_IN;
    float2 u01 = *(const float2*)(urow);
    float2 u23 = *(const float2*)(urow + 2);
    float2 u45 = *(const float2*)(urow + 4);
    pin_vgpr(u01.x); pin_vgpr(u01.y);
    pin_vgpr(u23.x); pin_vgpr(u23.y);
    pin_vgpr(u45.x); pin_vgpr(u45.y);

    // A 16x4 f32 layout: lanes 0-15 hold (K0,K1) in (V0,V1); lanes 16-31 hold (K2,K3).
    v2f aK03, aK47;
    aK03.x = hi ? u23.x : u01.x;          // WMMA #1: K = 0..3
    aK03.y = hi ? u23.y : u01.y;
    aK47.x = hi ? 0.0f : u45.x;           // WMMA #2: K = 4..7 (K6,K7 zero pad)
    aK47.y = hi ? 0.0f : u45.y;

    #pragma unroll
    for (int i = 0; i < 4; ++i) {
        const int n0 = (wave * 4 + i) * 16;

        // ---- B operand: w_in rows n0..n0+15 (B is KxN; w_in[n,k] is column n) ----
        // B 4x16 f32 layout: V0 = K0 | K2, V1 = K1 | K3 (lanes 0-15 | 16-31), N = lane%16.
        const float* wrow = w_in + (long)(n0 + m) * D_IN;
        float2 w01 = *(const float2*)(wrow);
        float2 w23 = *(const float2*)(wrow + 2);
        float2 w45 = *(const float2*)(wrow + 4);
        pin_vgpr(w01.x); pin_vgpr(w01.y);
        pin_vgpr(w23.x); pin_vgpr(w23.y);
        pin_vgpr(w45.x); pin_vgpr(w45.y);

        v2f bK03, bK47;
        bK03.x = hi ? w23.x : w01.x;
        bK03.y = hi ? w23.y : w01.y;
        bK47.x = hi ? 0.0f : w45.x;
        bK47.y = hi ? 0.0f : w45.y;

        v8f c = {};
        c = __builtin_amdgcn_wmma_f32_16x16x4_f32(false, aK03, false, bK03,
                                                  (short)0, c, false, false);
        c = __builtin_amdgcn_wmma_f32_16x16x4_f32(false, aK47, false, bK47,
                                                  (short)0, c, false, false);

        // ---- Epilogue: fuse noise add + relu + alpha scale + t==0 zeroing ----
        // C layout: VGPR j -> (M=j, N=lane) lanes 0-15; (M=j+8, N=lane-16) lanes 16-31.
        const int col = n0 + m;
        #pragma unroll
        for (int j = 0; j < 8; ++j) {
            const int  row = bt0 + j + (hi ? 8 : 0);
            const long idx = (long)row * N_DIM + col;
            float v = ALPHA_F * fmaxf(c[j] + noise[idx], 0.0f);
            if ((row & (T_DIM - 1)) == 0) v = 0.0f;   // states[:,0,:] stays zero
            out[idx] = v;
        }
    }
}

extern "C" void kernel_launch(void* const* d_in, const int* in_sizes, int n_in,
                              void* d_out, int out_size, void* d_ws, size_t ws_size,
                              hipStream_t stream) {
    // setup_inputs order: u [64,1024,6], w_rec [512,512] (unused), w_in [512,6], noise [64,1024,512]
    const float* u     = (const float*)d_in[0];
    const float* w_in  = (const float*)d_in[2];
    const float* noise = (const float*)d_in[3];
    float*       out   = (float*)d_out;

    const int BT = 64 * 1024;                 // 65536 bt rows
    const int blocks = BT / 16;               // one 16-row bt-tile per block -> 4096 blocks
    rnn_wmma_kernel<<<blocks, 256, 0, stream>>>(u, w_in, noise, out);
}